// SelfAttention_35373350650690
// MI455X (gfx1250) — compile-verified
//
#include <hip/hip_runtime.h>
#include <hip/hip_bf16.h>

// ---------------------------------------------------------------------------
// DGCNN EdgeConv pipeline for MI455X (gfx1250, wave32, WMMA).
//
// Factorization: W @ [x ; x_j - x] = (Wa - Wb)@x + Wb@x_j.
//   -> one dense GEMM per stage (WMMA f32 16x16x4) producing P (rows 0..M-1)
//      and Q (rows M..2M-1), then y[m,n,k] = P[m,n] + Q[m, idx[n,k]].
// InstanceNorm is a positive-scale affine per channel and lrelu is monotone,
// so max_k commutes with the normalization: keep per-channel (sum, sumsq)
// over all (n,k) and the per-(m,n) max of the raw y, normalize afterwards.
//
// GEMM: 2x2 register blocking per wave (32M x 32N, 4 accumulators), pointer-
// bumped addressing (no per-iteration 64-bit multiplies), K unrolled 4x ->
// 16 WMMAs per loop trip, 8 FLOP/byte from L2.
// ---------------------------------------------------------------------------

typedef __attribute__((ext_vector_type(2))) float v2f;
typedef __attribute__((ext_vector_type(8))) float v8f;

#define KNN_K 10
#define IN_EPS 1e-5f

__device__ __forceinline__ float lrelu(float y) { return y >= 0.0f ? y : 0.2f * y; }

__device__ __forceinline__ v8f wmma_f32(v2f a, v2f b, v8f c) {
    return __builtin_amdgcn_wmma_f32_16x16x4_f32(
        /*neg_a=*/false, a, /*neg_b=*/false, b,
        /*c_mod=*/(short)0, c, /*reuse_a=*/false, /*reuse_b=*/false);
}

// ---------------------------------------------------------------------------
// Prep: pts[b][n] = (x, y, z, |p|^2) from coords [B,3,N]
// ---------------------------------------------------------------------------
__global__ void prep_pts_kernel(const float* __restrict__ coords,
                                float4* __restrict__ pts, int N, int total) {
    int gid = blockIdx.x * 256 + threadIdx.x;
    if (gid >= total) return;
    int b = gid / N, n = gid - b * N;
    const float* cb = coords + (size_t)b * 3 * N;
    float x = cb[n], y = cb[N + n], z = cb[2 * N + n];
    pts[gid] = make_float4(x, y, z, x * x + y * y + z * z);
}

// W1 [128,256] -> W1cat [256,128]: rows 0..127 = W1a - W1b, rows 128.. = W1b
__global__ void prep_w1_kernel(const float* __restrict__ W1, float* __restrict__ Wc) {
    int gid = blockIdx.x * 256 + threadIdx.x;      // 128*128
    if (gid >= 128 * 128) return;
    int o = gid >> 7, c = gid & 127;
    float a = W1[o * 256 + c], b = W1[o * 256 + 128 + c];
    Wc[o * 128 + c] = a - b;
    Wc[(128 + o) * 128 + c] = b;
}

// W2 [256,256] -> W2cat [512,128]
__global__ void prep_w2_kernel(const float* __restrict__ W2, float* __restrict__ Wc) {
    int gid = blockIdx.x * 256 + threadIdx.x;      // 256*128
    if (gid >= 256 * 128) return;
    int o = gid >> 7, c = gid & 127;
    float a = W2[o * 256 + c], b = W2[o * 256 + 128 + c];
    Wc[o * 128 + c] = a - b;
    Wc[(256 + o) * 128 + c] = b;
}

// copy x0 (features) into rows 0..127 of the stacked X3 buffer [B,512,N]
__global__ void copy_x0_kernel(const float* __restrict__ feats,
                               float* __restrict__ X3, int N, int total) {
    int gid = blockIdx.x * 256 + threadIdx.x;      // B*128*N
    if (gid >= total) return;
    int n = gid % N, c = (gid / N) & 127, b = gid / (N * 128);
    X3[(size_t)b * 512 * N + (size_t)c * N + n] = feats[gid];
}

// ---------------------------------------------------------------------------
// KNN: per-thread row i, LDS-tiled candidates, unrolled 11-deep bubble insert.
// ---------------------------------------------------------------------------
__global__ __launch_bounds__(256) void knn_kernel(const float4* __restrict__ pts,
                                                  int* __restrict__ knn, int N) {
    __shared__ float4 tile[128];
    const int b = blockIdx.y;
    const int i = blockIdx.x * 256 + threadIdx.x;
    const float4 p = pts[(size_t)b * N + i];

    float bd[KNN_K + 1];
    int   bi[KNN_K + 1];
#pragma unroll
    for (int s = 0; s <= KNN_K; ++s) { bd[s] = 3.4e38f; bi[s] = -1; }

    for (int c0 = 0; c0 < N; c0 += 128) {
        __syncthreads();
        if (threadIdx.x < 128) tile[threadIdx.x] = pts[(size_t)b * N + c0 + threadIdx.x];
        __syncthreads();
        for (int j = 0; j < 128; ++j) {
            float4 q = tile[j];
            float d = p.w + q.w - 2.0f * (p.x * q.x + p.y * q.y + p.z * q.z);
            if (d < bd[KNN_K]) {                    // rare after warm-up
                float cd = d; int ci = c0 + j;
#pragma unroll
                for (int s = 0; s <= KNN_K; ++s) {  // bubble down, static indices
                    bool sw = cd < bd[s];
                    float td = sw ? bd[s] : cd; int ti = sw ? bi[s] : ci;
                    bd[s] = sw ? cd : bd[s];    bi[s] = sw ? ci : bi[s];
                    cd = td; ci = ti;
                }
            }
        }
    }
    int* o = knn + ((size_t)b * N + i) * KNN_K;
    int w = 0;
#pragma unroll
    for (int s = 0; s <= KNN_K; ++s)
        if (bi[s] != i && w < KNN_K) { o[w] = bi[s]; ++w; }  // drop self
}

// ---------------------------------------------------------------------------
// WMMA GEMM: D[b][M x N] = W[M x K] * X[b][K x N], all row-major, fp32.
// Block = 128 threads = 4 waves arranged 2x2; each wave owns a 32x32 D tile
// (2x2 register block of 16x16 WMMA accumulators). Block tile = 64M x 64N.
// Requires M % 64 == 0, N % 64 == 0, K % 16 == 0 -> uniform control flow,
// EXEC all ones around every WMMA. Addressing is pure pointer-bumping.
// ---------------------------------------------------------------------------
__global__ __launch_bounds__(128) void gemm_wmma_kernel(
        const float* __restrict__ W, const float* __restrict__ X,
        float* __restrict__ D, int N, int Kdim,
        long xBatchStride, long dBatchStride) {
    const int lane = threadIdx.x & 31;
    const int wave = threadIdx.x >> 5;
    const int half = lane >> 4;        // 0: lanes carry K,K+1 ; 1: K+2,K+3
    const int l16  = lane & 15;
    const int wm = wave >> 1, wn = wave & 1;
    const int m0 = blockIdx.y * 64 + wm * 32;
    const int n0 = blockIdx.x * 64 + wn * 32;
    const float* Xb = X + (size_t)blockIdx.z * xBatchStride;
    float*       Db = D + (size_t)blockIdx.z * dBatchStride;

    v8f c00 = {0.f,0.f,0.f,0.f,0.f,0.f,0.f,0.f};
    v8f c01 = c00, c10 = c00, c11 = c00;

    // Per-lane base pointers with the half-dependent K offset folded in once.
    const float* ar0 = W + (size_t)(m0 + l16) * Kdim + 2 * half;   // A rows m0..m0+15
    const float* ar1 = ar0 + (size_t)16 * Kdim;                    // A rows +16
    const float* xr  = Xb + (size_t)(2 * half) * N + n0 + l16;     // B cols n0..(+16)

    // Warm the panels into cache without branching in the hot loop.
    __builtin_prefetch(xr, 0, 1);
    __builtin_prefetch(ar0, 0, 1);

#pragma unroll 4
    for (int k = 0; k < Kdim; k += 4) {
        v2f a0, a1, b0, b1;
        a0.x = ar0[0];   a0.y = ar0[1];
        a1.x = ar1[0];   a1.y = ar1[1];
        b0.x = xr[0];    b0.y = xr[N];
        b1.x = xr[16];   b1.y = xr[16 + N];
        c00 = wmma_f32(a0, b0, c00);
        c01 = wmma_f32(a0, b1, c01);
        c10 = wmma_f32(a1, b0, c10);
        c11 = wmma_f32(a1, b1, c11);
        ar0 += 4; ar1 += 4; xr += (size_t)4 * N;
    }
#pragma unroll
    for (int v = 0; v < 8; ++v) {     // D vgpr v: M = v + 8*half, N = l16
        const int mr0 = m0 + v + 8 * half;
        Db[(size_t)mr0 * N + n0 + l16]             = c00[v];
        Db[(size_t)mr0 * N + n0 + 16 + l16]        = c01[v];
        Db[(size_t)(mr0 + 16) * N + n0 + l16]      = c10[v];
        Db[(size_t)(mr0 + 16) * N + n0 + 16 + l16] = c11[v];
    }
}

// ---------------------------------------------------------------------------
// Edge aggregation: y[m,n,k] = P[m,n] + Q[m, idx[n,k]];
// writes max_k y to outMax, accumulates (sum, sumsq) per (b,m) via atomics.
// grid = (N/256, M, B)
// ---------------------------------------------------------------------------
__global__ __launch_bounds__(256) void edge_agg_kernel(
        const float* __restrict__ PQ, const int* __restrict__ knn,
        float* __restrict__ outMax, float* __restrict__ stats,
        int M, int N, long outBatchStride) {
    __shared__ float rs[256], rs2[256];
    const int b = blockIdx.z, m = blockIdx.y;
    const int n = blockIdx.x * 256 + threadIdx.x;
    const float* P = PQ + ((size_t)b * 2 * M + m) * N;
    const float* Q = PQ + ((size_t)b * 2 * M + M + m) * N;
    const int* id = knn + ((size_t)b * N + n) * KNN_K;

    const float p = P[n];
    float mx = -3.4e38f, s = 0.f, s2 = 0.f;
#pragma unroll
    for (int k = 0; k < KNN_K; ++k) {
        float v = p + Q[id[k]];        // L2-resident gather
        mx = fmaxf(mx, v); s += v; s2 += v * v;
    }
    outMax[(size_t)b * outBatchStride + (size_t)m * N + n] = mx;

    rs[threadIdx.x] = s; rs2[threadIdx.x] = s2;
    __syncthreads();
    for (int off = 128; off > 0; off >>= 1) {
        if (threadIdx.x < off) {
            rs[threadIdx.x]  += rs[threadIdx.x + off];
            rs2[threadIdx.x] += rs2[threadIdx.x + off];
        }
        __syncthreads();
    }
    if (threadIdx.x == 0) {
        atomicAdd(&stats[((size_t)b * M + m) * 2 + 0], rs[0]);
        atomicAdd(&stats[((size_t)b * M + m) * 2 + 1], rs2[0]);
    }
}

// In-place InstanceNorm + lrelu using precomputed (sum, sumsq). cnt = N*K.
__global__ void in_norm_kernel(float* __restrict__ data,
                               const float* __restrict__ stats,
                               int M, int N, long batchStride, float invCnt,
                               int total) {
    int gid = blockIdx.x * 256 + threadIdx.x;
    if (gid >= total) return;
    int n = gid % N, m = (gid / N) % M, b = gid / (N * M);
    float s  = stats[((size_t)b * M + m) * 2 + 0];
    float s2 = stats[((size_t)b * M + m) * 2 + 1];
    float mu = s * invCnt;
    float var = fmaxf(s2 * invCnt - mu * mu, 0.f);
    float r = rsqrtf(var + IN_EPS);
    float* px = data + (size_t)b * batchStride + (size_t)m * N + n;
    *px = lrelu((*px - mu) * r);
}

// Final stage: per-(b,c) InstanceNorm over N (one block per row) + lrelu.
__global__ __launch_bounds__(256) void final_norm_kernel(
        const float* __restrict__ xin, float* __restrict__ out, int N) {
    __shared__ float rs[256], rs2[256];
    __shared__ float s_mu, s_r;
    const int b = blockIdx.y, c = blockIdx.x, C = gridDim.x;
    const float* row = xin + ((size_t)b * C + c) * N;
    float s = 0.f, s2 = 0.f;
    for (int n = threadIdx.x; n < N; n += 256) {
        float v = row[n]; s += v; s2 += v * v;
    }
    rs[threadIdx.x] = s; rs2[threadIdx.x] = s2;
    __syncthreads();
    for (int off = 128; off > 0; off >>= 1) {
        if (threadIdx.x < off) {
            rs[threadIdx.x]  += rs[threadIdx.x + off];
            rs2[threadIdx.x] += rs2[threadIdx.x + off];
        }
        __syncthreads();
    }
    if (threadIdx.x == 0) {
        float mu = rs[0] / (float)N;
        float var = fmaxf(rs2[0] / (float)N - mu * mu, 0.f);
        s_mu = mu; s_r = rsqrtf(var + IN_EPS);
    }
    __syncthreads();
    float mu = s_mu, r = s_r;
    float* orow = out + ((size_t)b * C + c) * N;
    for (int n = threadIdx.x; n < N; n += 256)
        orow[n] = lrelu((row[n] - mu) * r);
}

// ---------------------------------------------------------------------------
extern "C" void kernel_launch(void* const* d_in, const int* in_sizes, int n_in,
                              void* d_out, int out_size, void* d_ws, size_t ws_size,
                              hipStream_t stream) {
    constexpr int B = 4, C = 128, N = 4096;
    const float* coords = (const float*)d_in[0];   // [B,3,N]
    const float* feats  = (const float*)d_in[1];   // [B,128,N]
    const float* W1     = (const float*)d_in[2];   // [128,256]
    const float* W2     = (const float*)d_in[3];   // [256,256]
    const float* W3     = (const float*)d_in[4];   // [128,512]
    float* out = (float*)d_out;                    // [B,128,N]

    // workspace carve-up (256B aligned), ~74 MB total
    char* ws = (char*)d_ws;
    size_t off = 0;
    auto carve = [&](size_t bytes) -> void* {
        void* p = ws + off;
        off = (off + bytes + 255) & ~(size_t)255;
        return p;
    };
    float4* pts   = (float4*)carve((size_t)B * N * sizeof(float4));
    float* W1cat  = (float*)carve(256 * 128 * sizeof(float));
    float* W2cat  = (float*)carve(512 * 128 * sizeof(float));
    int*   knn    = (int*)carve((size_t)B * N * KNN_K * sizeof(int));
    float* PQ     = (float*)carve((size_t)B * 512 * N * sizeof(float)); // 32MB
    float* X3     = (float*)carve((size_t)B * 512 * N * sizeof(float)); // 32MB
    float* x3pre  = (float*)carve((size_t)B * 128 * N * sizeof(float)); // 8MB
    float* stats1 = (float*)carve((size_t)B * 128 * 2 * sizeof(float));
    float* stats2 = (float*)carve((size_t)B * 256 * 2 * sizeof(float));
    (void)ws_size; (void)in_sizes; (void)n_in; (void)out_size;

    hipMemsetAsync(stats1, 0, (size_t)B * 128 * 2 * sizeof(float), stream);
    hipMemsetAsync(stats2, 0, (size_t)B * 256 * 2 * sizeof(float), stream);

    // prep
    prep_pts_kernel<<<(B * N + 255) / 256, 256, 0, stream>>>(coords, pts, N, B * N);
    prep_w1_kernel<<<(128 * 128 + 255) / 256, 256, 0, stream>>>(W1, W1cat);
    prep_w2_kernel<<<(256 * 128 + 255) / 256, 256, 0, stream>>>(W2, W2cat);
    copy_x0_kernel<<<(B * 128 * N + 255) / 256, 256, 0, stream>>>(feats, X3, N, B * 128 * N);

    // KNN graph (shared by both EdgeConv stages)
    knn_kernel<<<dim3(N / 256, B), 256, 0, stream>>>(pts, knn, N);

    const float invCnt = 1.0f / (float)(N * KNN_K);

    // ---- stage 1: PQ1[256xN] = W1cat @ x0 ------------------------------
    gemm_wmma_kernel<<<dim3(N / 64, 256 / 64, B), 128, 0, stream>>>(
        W1cat, feats, PQ, N, C, (long)C * N, (long)256 * N);
    edge_agg_kernel<<<dim3(N / 256, 128, B), 256, 0, stream>>>(
        PQ, knn, X3 + (size_t)128 * N, stats1, 128, N, (long)512 * N);
    in_norm_kernel<<<(B * 128 * N + 255) / 256, 256, 0, stream>>>(
        X3 + (size_t)128 * N, stats1, 128, N, (long)512 * N, invCnt, B * 128 * N);

    // ---- stage 2: PQ2[512xN] = W2cat @ x1 ------------------------------
    gemm_wmma_kernel<<<dim3(N / 64, 512 / 64, B), 128, 0, stream>>>(
        W2cat, X3 + (size_t)128 * N, PQ, N, C, (long)512 * N, (long)512 * N);
    edge_agg_kernel<<<dim3(N / 256, 256, B), 256, 0, stream>>>(
        PQ, knn, X3 + (size_t)256 * N, stats2, 256, N, (long)512 * N);
    in_norm_kernel<<<(B * 256 * N + 255) / 256, 256, 0, stream>>>(
        X3 + (size_t)256 * N, stats2, 256, N, (long)512 * N, invCnt, B * 256 * N);

    // ---- fuse: x3pre[128xN] = W3 @ [x0;x1;x2], IN over N, lrelu --------
    gemm_wmma_kernel<<<dim3(N / 64, 128 / 64, B), 128, 0, stream>>>(
        W3, X3, x3pre, N, 512, (long)512 * N, (long)128 * N);
    final_norm_kernel<<<dim3(128, B), 256, 0, stream>>>(x3pre, out, N);
}